// TransitionDownModule_70196945486298
// MI455X (gfx1250) — compile-verified
//
#include <hip/hip_runtime.h>
#include <math.h>

typedef __attribute__((ext_vector_type(2))) float v2f;
typedef __attribute__((ext_vector_type(8))) float v8f;
typedef int v4i __attribute__((vector_size(4 * sizeof(int))));

#define AS3 __attribute__((address_space(3)))
#define AS1 __attribute__((address_space(1)))

#define B_   8
#define N_   8192
#define D_   64
#define N2_  2048
#define K_   16
#define MT   16                    // queries per block
#define NBLK (B_ * N2_ / MT)       // 1024
#define CHUNK 128
#define QSTR 68                    // padded row stride (floats) to dodge bank conflicts
#define DSTR 132

// ---------------------------------------------------------------------------
// CDNA5 async global->LDS copy (ASYNCcnt path), 16B per lane.
// ---------------------------------------------------------------------------
__device__ __forceinline__ void async_copy_b128(const float* g, float* l) {
#if __has_builtin(__builtin_amdgcn_global_load_async_to_lds_b128)
  __builtin_amdgcn_global_load_async_to_lds_b128((AS1 v4i*)(void*)g,
                                                 (AS3 v4i*)(void*)l, 0, 0);
#else
  asm volatile("global_load_async_to_lds_b128 %0, %1, off"
               :: "v"((AS3 v4i*)(void*)l), "v"((AS1 v4i*)(void*)g) : "memory");
#endif
}

__device__ __forceinline__ void wait_async0() {
#if __has_builtin(__builtin_amdgcn_s_wait_asynccnt)
  __builtin_amdgcn_s_wait_asynccnt(0);
#else
  asm volatile("s_wait_asynccnt 0x0" ::: "memory");
#endif
}

// ---------------------------------------------------------------------------
// Kernel 0: per-point squared norms |x_n|^2 -> ws (needed for distance score)
// ---------------------------------------------------------------------------
__global__ __launch_bounds__(256) void norms_kernel(const float* __restrict__ x,
                                                    float* __restrict__ xn) {
  const int i = blockIdx.x * 256 + threadIdx.x;      // 0 .. B*N-1
  const float* r = x + (size_t)i * D_;
  float s = 0.f;
  for (int c4 = 0; c4 < 16; ++c4) {
    const float4 f = *(const float4*)(r + c4 * 4);
    s += f.x * f.x + f.y * f.y + f.z * f.z + f.w * f.w;
  }
  xn[i] = s;
}

// ---------------------------------------------------------------------------
// Kernel 1: fused distance GEMM (f32 WMMA 16x16x4) + top-k(16) selection.
// One block = one batch b + one tile of 16 queries. 8 waves, each wave owns a
// 16-candidate column tile per chunk of 128 candidates.
// score(n) = |x_n|^2 - 2 * <q_m, x_n>   (|q|^2 dropped: rank-invariant per row)
// ---------------------------------------------------------------------------
__global__ __launch_bounds__(256) void knn_topk_kernel(const float* __restrict__ x,
                                                       const float* __restrict__ xn,
                                                       int* __restrict__ idx_out) {
  __shared__ float lds_q[MT * QSTR];        // 16 x 64 query tile
  __shared__ float lds_x[CHUNK * QSTR];     // 128 x 64 candidate chunk
  __shared__ float lds_dist[MT * DSTR];     // 16 x 128 scores
  __shared__ float lds_bd[MT * K_];         // running best distances (sorted asc)
  __shared__ int   lds_bi[MT * K_];         // running best indices

  const int t    = threadIdx.x;
  const int lane = t & 31;
  const int w    = t >> 5;
  const int hv   = lane >> 4;               // lane half (wave32)
  const int l15  = lane & 15;
  const int blk  = blockIdx.x;
  const int b    = blk >> 7;                // / (N2/MT)
  const int m0   = (blk & 127) * MT;

  const float* xb  = x + (size_t)b * N_ * D_;
  const float* xng = xn + (size_t)b * N_;

  // stage query tile via async copy (queries are x rows (m0+r)*4)
  {
    const int r = t >> 4, c4 = (t & 15) * 4;
    async_copy_b128(xb + (size_t)((m0 + r) * 4) * D_ + c4, &lds_q[r * QSTR + c4]);
  }
  // init best lists
  {
    const int r = t >> 4, s = t & 15;
    lds_bd[r * K_ + s] = 3.0e38f;
    lds_bi[r * K_ + s] = 0;
  }
  wait_async0();
  __syncthreads();

  // Build the A-matrix fragments once: 16 K-steps of (K=4) each; per lane the
  // (row = l15, K pair selected by lane half) layout of V_WMMA_F32_16X16X4_F32.
  v2f a[16];
  for (int kk = 0; kk < 16; ++kk) {
    a[kk].x = lds_q[l15 * QSTR + kk * 4 + 2 * hv];
    a[kk].y = lds_q[l15 * QSTR + kk * 4 + 2 * hv + 1];
  }

  for (int ch = 0; ch < N_ / CHUNK; ++ch) {
    const int n0 = ch * CHUNK;

    // stage 128x64 chunk straight into LDS (8 x B128 per thread, async path)
    for (int u = 0; u < 8; ++u) {
      const int v = t * 8 + u;
      const int row = v >> 4, c4 = (v & 15) * 4;
      async_copy_b128(xb + (size_t)(n0 + row) * D_ + c4, &lds_x[row * QSTR + c4]);
    }
    // candidate norm for this wave's column (global scalar, overlaps async)
    const float xnv = xng[n0 + w * 16 + l15];
    if (ch + 1 < N_ / CHUNK && t < CHUNK)
      __builtin_prefetch(xb + (size_t)(n0 + CHUNK + t) * D_, 0, 0);
    wait_async0();
    __syncthreads();                        // [B] chunk visible to all waves

    // WMMA: 16 steps of f32 16x16x4 accumulate the 16x16 dot-product tile
    v8f c = {};
    {
      const int jb = w * 16;
      for (int kk = 0; kk < 16; ++kk) {
        v2f bv;
        bv.x = lds_x[(jb + l15) * QSTR + kk * 4 + 2 * hv];
        bv.y = lds_x[(jb + l15) * QSTR + kk * 4 + 2 * hv + 1];
        c = __builtin_amdgcn_wmma_f32_16x16x4_f32(false, a[kk], false, bv,
                                                  (short)0, c, false, false);
      }
    }

    // scores into LDS (C layout: VGPR r -> row r + 8*half, col = lane&15)
    {
      const int j = w * 16 + l15;
      for (int r = 0; r < 8; ++r)
        lds_dist[(r + 8 * hv) * DSTR + j] = xnv - 2.0f * c[r];
    }
    __syncthreads();                        // [C] scores visible, lds_x free

    // top-k merge: lanes 0,1 of each wave each own one row (16 rows total)
    if (lane < 2) {
      const int row = 2 * w + lane;
      float thr = lds_bd[row * K_ + (K_ - 1)];
      for (int j = 0; j < CHUNK; ++j) {
        const float dv = lds_dist[row * DSTR + j];
        if (dv < thr) {
          int p = K_ - 1;
          while (p > 0 && lds_bd[row * K_ + p - 1] > dv) {
            lds_bd[row * K_ + p] = lds_bd[row * K_ + p - 1];
            lds_bi[row * K_ + p] = lds_bi[row * K_ + p - 1];
            --p;
          }
          lds_bd[row * K_ + p] = dv;
          lds_bi[row * K_ + p] = n0 + j;
          thr = lds_bd[row * K_ + (K_ - 1)];
        }
      }
    }
    // no barrier needed here: next chunk writes lds_x (not read here) and the
    // next dist write happens only after barrier [B] of the next iteration.
  }
  __syncthreads();
  {
    const int r = t >> 4, s = t & 15;
    idx_out[(size_t)(b * N2_ + m0 + r) * K_ + s] = lds_bi[r * K_ + s];
  }
}

// ---------------------------------------------------------------------------
// Kernel 2/4: gather neighbors + (16x64)x(64x64) WMMA GEMM (+bias).
// mode 0: accumulate per-channel sum / sumsq -> per-block partials.
// mode 1: BN-normalize with precomputed stats, ReLU, max over k, write out.
// ---------------------------------------------------------------------------
__global__ __launch_bounds__(256) void gather_gemm_kernel(
    const float* __restrict__ x, const float* __restrict__ W,
    const float* __restrict__ bias, const float* __restrict__ gamma,
    const float* __restrict__ beta, const int* __restrict__ idx,
    float* __restrict__ part, const float* __restrict__ stats,
    float* __restrict__ out, int mode) {
  __shared__ float lds_w[D_ * QSTR];          // 64 x 64 W (row o, col c)
  __shared__ float lds_a[8 * MT * QSTR];      // per-wave slab: 16 rows x 64 (A, then y)
  __shared__ int   lds_idx[MT * K_];
  __shared__ float lds_s1[8 * D_];
  __shared__ float lds_s2[8 * D_];

  const int t    = threadIdx.x;
  const int lane = t & 31;
  const int w    = t >> 5;
  const int hv   = lane >> 4;
  const int l15  = lane & 15;
  const int blk  = blockIdx.x;
  const int b    = blk >> 7;
  const int m0   = (blk & 127) * MT;

  const float* xb = x + (size_t)b * N_ * D_;

  // stage W via async: y[.,o] = sum_c x2[.,c] * W[o,c]; B-matrix is W^T ([o][c])
  for (int u = 0; u < 4; ++u) {
    const int v = t * 4 + u;
    const int o = v >> 4, c4 = (v & 15) * 4;
    async_copy_b128(W + (size_t)o * D_ + c4, &lds_w[o * QSTR + c4]);
  }
  lds_idx[t] = idx[(size_t)(b * N2_ + m0) * K_ + t];
  lds_s1[t] = 0.f; lds_s1[t + 256] = 0.f;
  lds_s2[t] = 0.f; lds_s2[t + 256] = 0.f;
  wait_async0();
  __syncthreads();

  const int slab = w * MT * QSTR;
  for (int iter = 0; iter < 2; ++iter) {
    const int q = w + iter * 8;             // this wave's query within the tile

    // gather this query's 16 neighbor rows into the wave's slab (async)
    for (int u = 0; u < 8; ++u) {
      const int v = lane * 8 + u;
      const int s = v >> 4, c4 = (v & 15) * 4;
      const int gi = lds_idx[q * K_ + s];
      async_copy_b128(xb + (size_t)gi * D_ + c4, &lds_a[slab + s * QSTR + c4]);
    }
    wait_async0();
    __syncthreads();

    // A fragments (16 neighbor rows x 64 channels)
    v2f a[16];
    for (int kk = 0; kk < 16; ++kk) {
      a[kk].x = lds_a[slab + l15 * QSTR + kk * 4 + 2 * hv];
      a[kk].y = lds_a[slab + l15 * QSTR + kk * 4 + 2 * hv + 1];
    }

    // 4 column tiles of 16 output channels each
    v8f yt[4];
    for (int ct = 0; ct < 4; ++ct) {
      v8f c = {};
      for (int kk = 0; kk < 16; ++kk) {
        v2f bv;
        bv.x = lds_w[(ct * 16 + l15) * QSTR + kk * 4 + 2 * hv];
        bv.y = lds_w[(ct * 16 + l15) * QSTR + kk * 4 + 2 * hv + 1];
        c = __builtin_amdgcn_wmma_f32_16x16x4_f32(false, a[kk], false, bv,
                                                  (short)0, c, false, false);
      }
      const float bo = bias[ct * 16 + l15];
      for (int r = 0; r < 8; ++r) c[r] += bo;
      yt[ct] = c;
    }

    // scatter y into the slab (overwrites A; A already in registers)
    for (int ct = 0; ct < 4; ++ct) {
      const int o = ct * 16 + l15;
      for (int r = 0; r < 8; ++r)
        lds_a[slab + (r + 8 * hv) * QSTR + o] = yt[ct][r];
    }
    __syncthreads();

    if (mode == 0) {
      for (int h = 0; h < 2; ++h) {
        const int o = lane + 32 * h;
        float s1 = 0.f, s2 = 0.f;
        for (int s = 0; s < MT; ++s) {
          const float v = lds_a[slab + s * QSTR + o];
          s1 += v; s2 += v * v;
        }
        lds_s1[w * D_ + o] += s1;           // wave-private: no races
        lds_s2[w * D_ + o] += s2;
      }
    } else {
      for (int h = 0; h < 2; ++h) {
        const int o  = lane + 32 * h;
        const float mn = stats[2 * o];
        const float iv = stats[2 * o + 1];
        const float g  = gamma[o];
        const float be = beta[o];
        float mx = 0.f;                     // max(0, max_s(.)) == max_s relu(.)
        for (int s = 0; s < MT; ++s) {
          const float v  = lds_a[slab + s * QSTR + o];
          const float tn = g * (v - mn) * iv + be;
          mx = fmaxf(mx, tn);
        }
        out[(size_t)(b * N2_ + m0 + q) * D_ + o] = mx;
      }
    }
    __syncthreads();                        // slab reuse fence for next iter
  }

  if (mode == 0 && t < D_) {
    float s1 = 0.f, s2 = 0.f;
    for (int ww = 0; ww < 8; ++ww) {
      s1 += lds_s1[ww * D_ + t];
      s2 += lds_s2[ww * D_ + t];
    }
    part[(size_t)(blk * D_ + t) * 2 + 0] = s1;
    part[(size_t)(blk * D_ + t) * 2 + 1] = s2;
  }
}

// ---------------------------------------------------------------------------
// Kernel 3: deterministic f64 reduction of per-block partials -> mean, rstd
// ---------------------------------------------------------------------------
__global__ void bn_stats_kernel(const float* __restrict__ part,
                                float* __restrict__ stats) {
  const int o = threadIdx.x;
  if (o < D_) {
    double s1 = 0.0, s2 = 0.0;
    for (int p = 0; p < NBLK; ++p) {
      s1 += (double)part[(size_t)(p * D_ + o) * 2 + 0];
      s2 += (double)part[(size_t)(p * D_ + o) * 2 + 1];
    }
    const double cnt  = (double)B_ * N2_ * K_;
    const double mean = s1 / cnt;
    const double var  = s2 / cnt - mean * mean;
    stats[2 * o]     = (float)mean;
    stats[2 * o + 1] = (float)(1.0 / sqrt(var + 1e-5));
  }
}

extern "C" void kernel_launch(void* const* d_in, const int* in_sizes, int n_in,
                              void* d_out, int out_size, void* d_ws, size_t ws_size,
                              hipStream_t stream) {
  (void)in_sizes; (void)n_in; (void)out_size; (void)ws_size;
  const float* x     = (const float*)d_in[0];
  // d_in[1] = p (unused by the reference)
  const float* W     = (const float*)d_in[2];
  const float* bias  = (const float*)d_in[3];
  const float* gamma = (const float*)d_in[4];
  const float* beta  = (const float*)d_in[5];
  float* out = (float*)d_out;

  char* ws = (char*)d_ws;
  int*   idx   = (int*)ws;                          // B*N2*K ints   (1 MB)
  float* part  = (float*)(ws + 2097152);            // NBLK*64*2 f32 (512 KB)
  float* stats = (float*)(ws + 2621440);            // 64*2 f32
  float* xn    = (float*)(ws + 3145728);            // B*N f32      (256 KB)

  norms_kernel<<<(B_ * N_) / 256, 256, 0, stream>>>(x, xn);
  knn_topk_kernel<<<NBLK, 256, 0, stream>>>(x, xn, idx);
  gather_gemm_kernel<<<NBLK, 256, 0, stream>>>(x, W, bias, gamma, beta, idx,
                                               part, stats, out, 0);
  bn_stats_kernel<<<1, 64, 0, stream>>>(part, stats);
  gather_gemm_kernel<<<NBLK, 256, 0, stream>>>(x, W, bias, gamma, beta, idx,
                                               part, stats, out, 1);
}